// RandomAugment_61177514164843
// MI455X (gfx1250) — compile-verified
//
#include <hip/hip_runtime.h>
#include <math.h>

// CDNA5 / gfx1250. wave32. Uses V_WMMA_F32_16X16X4_F32 for the pairwise
// distance GEMM, LDS staging of the transformed point table, and shfl
// butterflies for the row-min reduction.

typedef float v2f __attribute__((ext_vector_type(2)));
typedef float v8f __attribute__((ext_vector_type(8)));

#define EPS    1.0f
#define BLK    256        // 8 waves per block
#define CHUNK  2048       // points staged into LDS per chunk (32 KB of float4)

__global__ void copy_p_kernel(const float* __restrict__ p,
                              float* __restrict__ out, int n3) {
  int i = blockIdx.x * blockDim.x + threadIdx.x;
  if (i < n3) out[i] = p[i];
}

__global__ __launch_bounds__(BLK) void
augment_kernel(const float* __restrict__ p, const float* __restrict__ u,
               float* __restrict__ out, int np, int ms) {
  __shared__ float4 qs[CHUNK];   // staged q[n] = (-2p0,-2p1,-2p2,||p||^2)

  const int tid  = threadIdx.x;
  const int lane = tid & 31;
  const int wave = tid >> 5;
  const int row  = lane & 15;    // M-row (for A) / N-col (for B) index in tile
  const int half = lane >> 4;    // 0: K=0,1  1: K=2,3

  // ---------- Pass 1: bounding box of p (block-local; p is L2-resident) ----
  float mn0 = 3.4e38f, mn1 = 3.4e38f, mn2 = 3.4e38f;
  float mx0 = -3.4e38f, mx1 = -3.4e38f, mx2 = -3.4e38f;
  for (int i = tid; i < np; i += BLK) {
    float a0 = p[i * 3 + 0], a1 = p[i * 3 + 1], a2 = p[i * 3 + 2];
    mn0 = fminf(mn0, a0); mx0 = fmaxf(mx0, a0);
    mn1 = fminf(mn1, a1); mx1 = fmaxf(mx1, a1);
    mn2 = fminf(mn2, a2); mx2 = fmaxf(mx2, a2);
  }
  float* red = (float*)qs;  // reuse LDS for the reduction
  red[0 * BLK + tid] = mn0; red[1 * BLK + tid] = mn1; red[2 * BLK + tid] = mn2;
  red[3 * BLK + tid] = mx0; red[4 * BLK + tid] = mx1; red[5 * BLK + tid] = mx2;
  __syncthreads();
  for (int s = BLK / 2; s > 0; s >>= 1) {
    if (tid < s) {
      red[0 * BLK + tid] = fminf(red[0 * BLK + tid], red[0 * BLK + tid + s]);
      red[1 * BLK + tid] = fminf(red[1 * BLK + tid], red[1 * BLK + tid + s]);
      red[2 * BLK + tid] = fminf(red[2 * BLK + tid], red[2 * BLK + tid + s]);
      red[3 * BLK + tid] = fmaxf(red[3 * BLK + tid], red[3 * BLK + tid + s]);
      red[4 * BLK + tid] = fmaxf(red[4 * BLK + tid], red[4 * BLK + tid + s]);
      red[5 * BLK + tid] = fmaxf(red[5 * BLK + tid], red[5 * BLK + tid + s]);
    }
    __syncthreads();
  }
  mn0 = red[0 * BLK]; mn1 = red[1 * BLK]; mn2 = red[2 * BLK];
  float sc0 = red[3 * BLK] - mn0;
  float sc1 = red[4 * BLK] - mn1;
  float sc2 = red[5 * BLK] - mn2;
  __syncthreads();  // everyone done reading `red` before staging overwrites qs

  // ---------- this lane's sample s (row = lane%16 of this wave's M tile) ---
  int g = blockIdx.x * 128 + wave * 16 + row;          // sample index < ms
  float s0 = u[g * 3 + 0] * sc0 + mn0;
  float s1 = u[g * 3 + 1] * sc1 + mn1;
  float s2 = u[g * 3 + 2] * sc2 + mn2;

  // A matrix 16x4 (f32): lanes 0-15 -> K=0,1 ; lanes 16-31 -> K=2,3
  // A row m = [s0, s1, s2, 1]
  v2f amat;
  amat.x = half ? s2 : s0;
  amat.y = half ? 1.0f : s1;

  v8f minD;
#pragma unroll
  for (int i = 0; i < 8; ++i) minD[i] = 3.4e38f;

  // ---------- main loop: WMMA over all N tiles, chunked through LDS --------
  const int nchunks = np / CHUNK;
  for (int c = 0; c < nchunks; ++c) {
#pragma unroll
    for (int i = 0; i < CHUNK / BLK; ++i) {   // stage + transform p -> q
      int idx = c * CHUNK + i * BLK + tid;
      float x = p[idx * 3 + 0], y = p[idx * 3 + 1], z = p[idx * 3 + 2];
      qs[i * BLK + tid] = make_float4(-2.f * x, -2.f * y, -2.f * z,
                                      x * x + y * y + z * z);
    }
    __syncthreads();
    const float2* qf2 = (const float2*)qs;
#pragma unroll 4
    for (int t = 0; t < CHUNK / 16; ++t) {
      // B matrix 4x16: col n = [-2p0, -2p1, -2p2, ||p_n||^2]^T
      float2 bq = qf2[(((t << 4) | row) << 1) + half];
      v2f bmat; bmat.x = bq.x; bmat.y = bq.y;
      v8f cz = {};
      // D[m][n] = ||p_n||^2 - 2 s_m . p_n
      v8f d = __builtin_amdgcn_wmma_f32_16x16x4_f32(
          false, amat, false, bmat, (short)0, cz, false, false);
#pragma unroll
      for (int i = 0; i < 8; ++i) minD[i] = fminf(minD[i], d[i]);
    }
    __syncthreads();
  }

  // ---------- reduce min over the 16 N-columns (lanes within 16-groups) ----
#pragma unroll
  for (int off = 1; off < 16; off <<= 1) {
#pragma unroll
    for (int i = 0; i < 8; ++i)
      minD[i] = fminf(minD[i], __shfl_xor(minD[i], off, 32));
  }
  // entry r of lane-group holds row m = r + 8*(group); pick this lane's row
  int r = row & 7;
  float xm = minD[0];
#pragma unroll
  for (int i = 1; i < 8; ++i) xm = (r == i) ? minD[i] : xm;
  int src = (row < 8) ? row : (row + 16);  // rows 8-15 live in the upper half
  xm = __shfl(xm, src, 32);

  // ---------- finish: add ||s||^2, clip, sqrt, threshold, write ------------
  float d2 = fmaxf(xm + (s0 * s0 + s1 * s1 + s2 * s2), 0.0f);
  bool keep = sqrtf(d2) > EPS;
  if (half == 0) {                         // lanes 16-31 are duplicates
    float o0 = keep ? s0 : 0.0f;
    float o1 = keep ? s1 : 0.0f;
    float o2 = keep ? s2 : 0.0f;
    out[(np + g) * 3 + 0] = o0;
    out[(np + g) * 3 + 1] = o1;
    out[(np + g) * 3 + 2] = o2;
  }
}

extern "C" void kernel_launch(void* const* d_in, const int* in_sizes, int n_in,
                              void* d_out, int out_size, void* d_ws, size_t ws_size,
                              hipStream_t stream) {
  (void)n_in; (void)out_size; (void)d_ws; (void)ws_size;
  const float* p = (const float*)d_in[0];   // [np, 3] fp32
  const float* u = (const float*)d_in[1];   // [ms, 3] fp32
  float* out = (float*)d_out;               // [np + ms, 3] fp32
  int np = in_sizes[0] / 3;                 // 8192
  int ms = in_sizes[1] / 3;                 // 32768
  int n3 = np * 3;

  copy_p_kernel<<<(n3 + BLK - 1) / BLK, BLK, 0, stream>>>(p, out, n3);
  // 128 samples (8 M-tiles of 16) per block
  augment_kernel<<<ms / 128, BLK, 0, stream>>>(p, u, out, np, ms);
}